// HANN_75780402970817
// MI455X (gfx1250) — compile-verified
//
#include <hip/hip_runtime.h>
#include <hip/hip_bf16.h>
#include <math.h>

// ---------- problem constants (from reference setup_inputs) ----------
#define S_   256
#define L_   256
#define E_   512
#define N_   65536         // S_*L_ hyper vertices
#define KH_  4             // hyperedges per vertex
#define T_   8192
#define A_   8192
#define D_   256
#define NT_  40            // trigger classes
#define NR_  20            // role classes

typedef __attribute__((ext_vector_type(2))) float v2f;
typedef __attribute__((ext_vector_type(8))) float v8f;

#define WMMA_F32(a, b, c) \
    __builtin_amdgcn_wmma_f32_16x16x4_f32(false, (a), false, (b), (short)0, (c), false, false)

// ---------------------------------------------------------------
// 1) embedding gather + positional row (odd dims get +1.0)
// ---------------------------------------------------------------
__global__ void hann_emb_pe(const int* __restrict__ x, const float* __restrict__ emb,
                            float* __restrict__ h0) {
    int i = blockIdx.x * blockDim.x + threadIdx.x;          // over S*L*D
    int node = i >> 8;                                      // /D
    int d    = i & (D_ - 1);
    float pe = (d & 1) ? 1.0f : 0.0f;
    h0[i] = emb[(size_t)x[node] * D_ + d] + pe;
}

// ---------------------------------------------------------------
// 2) fp32 WMMA GEMM: C[M,N] = A[M,K] * B[K,N]
//    one wave per 16x64 C strip (4 tiles share one A fragment)
// ---------------------------------------------------------------
__global__ void hann_gemm_f32(const float* __restrict__ A, const float* __restrict__ B,
                              float* __restrict__ C, int M, int N, int K) {
    int stripsN = N >> 6;                       // 64-wide strips
    int tile = blockIdx.x;
    int tm = tile / stripsN, tn = tile % stripsN;
    int lane = threadIdx.x & 31;
    int mrow = lane & 15;                       // A row within tile / B,C col within tile
    int hlf  = lane >> 4;                       // 0: K pair {0,1}, 1: K pair {2,3}

    const float* Arow = A + (size_t)(tm * 16 + mrow) * K;
    int nbase = tn * 64 + mrow;

    v8f acc0 = {}, acc1 = {}, acc2 = {}, acc3 = {};
    for (int k = 0; k < K; k += 4) {
        int ka = k + 2 * hlf;
        v2f a;
        a.x = Arow[ka];
        a.y = Arow[ka + 1];
        const float* Bk0 = B + (size_t)ka * N + nbase;
        const float* Bk1 = B + (size_t)(ka + 1) * N + nbase;
        v2f b0, b1, b2, b3;
        b0.x = Bk0[0];  b0.y = Bk1[0];
        b1.x = Bk0[16]; b1.y = Bk1[16];
        b2.x = Bk0[32]; b2.y = Bk1[32];
        b3.x = Bk0[48]; b3.y = Bk1[48];
        acc0 = WMMA_F32(a, b0, acc0);
        acc1 = WMMA_F32(a, b1, acc1);
        acc2 = WMMA_F32(a, b2, acc2);
        acc3 = WMMA_F32(a, b3, acc3);
    }
    // D layout: VGPR r -> M = r + hlf*8, N = mrow
    #pragma unroll
    for (int r = 0; r < 8; ++r) {
        size_t row = (size_t)(tm * 16 + r + hlf * 8) * N;
        C[row + nbase + 0 ] = acc0[r];
        C[row + nbase + 16] = acc1[r];
        C[row + nbase + 32] = acc2[r];
        C[row + nbase + 48] = acc3[r];
    }
}

// ---------------------------------------------------------------
// 3) GCN normalization helpers
// ---------------------------------------------------------------
__global__ void hann_deg_init(float* __restrict__ deg) {
    int i = blockIdx.x * blockDim.x + threadIdx.x;          // S*L
    deg[i] = 1.0f;                                          // self loop
}
__global__ void hann_deg_accum(const int* __restrict__ edst, float* __restrict__ deg) {
    int i = blockIdx.x * blockDim.x + threadIdx.x;          // S*E
    int s = i / E_;
    atomicAdd(&deg[s * L_ + edst[i]], 1.0f);
}
__global__ void hann_dinv(float* __restrict__ deg) {
    int i = blockIdx.x * blockDim.x + threadIdx.x;          // S*L
    deg[i] = rsqrtf(deg[i]);                                // deg >= 1 always
}

// self-loop term: out = hW * dinv^2
__global__ void hann_agg_init(const float* __restrict__ hW, const float* __restrict__ dinv,
                              float* __restrict__ out) {
    int i = blockIdx.x * blockDim.x + threadIdx.x;          // S*L*D
    float di = dinv[i >> 8];
    out[i] = hW[i] * di * di;
}
// edge scatter-add: one block per edge, D threads
__global__ void hann_agg_edges(const float* __restrict__ hW, const float* __restrict__ dinv,
                               const int* __restrict__ esrc, const int* __restrict__ edst,
                               float* __restrict__ out) {
    int e = blockIdx.x;                                     // S*E
    int s = e / E_;
    int src = s * L_ + esrc[e];
    int dst = s * L_ + edst[e];
    float nrm = dinv[src] * dinv[dst];
    int d = threadIdx.x;
    atomicAdd(&out[(size_t)dst * D_ + d], hW[(size_t)src * D_ + d] * nrm);
}
__global__ void hann_bias_relu(float* __restrict__ out, const float* __restrict__ b) {
    int i = blockIdx.x * blockDim.x + threadIdx.x;          // S*L*D
    out[i] = fmaxf(out[i] + b[i & (D_ - 1)], 0.0f);
}

// ---------------------------------------------------------------
// 4) sentence summary: max over L per (s, d)
// ---------------------------------------------------------------
__global__ void hann_sent_max(const float* __restrict__ h2, float* __restrict__ sents) {
    int s = blockIdx.x;
    int d = threadIdx.x;
    float mx = -INFINITY;
    const float* base = h2 + (size_t)s * L_ * D_ + d;
    for (int l = 0; l < L_; ++l) mx = fmaxf(mx, base[(size_t)l * D_]);
    sents[s * D_ + d] = mx;
}

// hypergraph: m[v] = max over KH_ incident sentence summaries
__global__ void hann_hyper_max(const float* __restrict__ sents, const int* __restrict__ he,
                               float* __restrict__ mbuf) {
    int v = blockIdx.x;
    int d = threadIdx.x;
    float mx = -INFINITY;
    #pragma unroll
    for (int k = 0; k < KH_; ++k) {
        int s = he[v * KH_ + k];
        mx = fmaxf(mx, sents[s * D_ + d]);
    }
    mbuf[(size_t)v * D_ + d] = mx;
}

// ---------------------------------------------------------------
// 5) ffnn1 WMMA GEMM with gathered A rows and scattered C rows:
//    out2[hv[r]] = concat(outs[hv[r]], m[r]) @ W1 + b1    (K=512, N=256)
//    one wave per 16x64 strip
// ---------------------------------------------------------------
__global__ void hann_ffnn1_wmma(const float* __restrict__ outs, const float* __restrict__ mbuf,
                                const int* __restrict__ hv, const float* __restrict__ W1,
                                const float* __restrict__ b1, float* __restrict__ out2) {
    const int N = D_, K = 2 * D_;
    int stripsN = N >> 6;
    int tile = blockIdx.x;
    int tm = tile / stripsN, tn = tile % stripsN;
    int lane = threadIdx.x & 31;
    int mrow = lane & 15;
    int hlf  = lane >> 4;

    int arow = tm * 16 + mrow;
    int g = hv[arow];
    const float* Alo = outs + (size_t)g * D_;       // K in [0,256)
    const float* Ahi = mbuf + (size_t)arow * D_;    // K in [256,512)
    int nbase = tn * 64 + mrow;

    v8f acc0 = {}, acc1 = {}, acc2 = {}, acc3 = {};
    for (int k = 0; k < K; k += 4) {
        int ka = k + 2 * hlf;
        v2f a;
        if (ka < D_) { a.x = Alo[ka];      a.y = Alo[ka + 1]; }
        else         { a.x = Ahi[ka - D_]; a.y = Ahi[ka - D_ + 1]; }
        const float* Bk0 = W1 + (size_t)ka * N + nbase;
        const float* Bk1 = W1 + (size_t)(ka + 1) * N + nbase;
        v2f b0, b1v, b2, b3;
        b0.x  = Bk0[0];  b0.y  = Bk1[0];
        b1v.x = Bk0[16]; b1v.y = Bk1[16];
        b2.x  = Bk0[32]; b2.y  = Bk1[32];
        b3.x  = Bk0[48]; b3.y  = Bk1[48];
        acc0 = WMMA_F32(a, b0,  acc0);
        acc1 = WMMA_F32(a, b1v, acc1);
        acc2 = WMMA_F32(a, b2,  acc2);
        acc3 = WMMA_F32(a, b3,  acc3);
    }
    #pragma unroll
    for (int r = 0; r < 8; ++r) {
        int row = tm * 16 + r + hlf * 8;
        size_t gs = (size_t)hv[row] * D_;
        out2[gs + nbase + 0 ] = acc0[r] + b1[nbase + 0 ];
        out2[gs + nbase + 16] = acc1[r] + b1[nbase + 16];
        out2[gs + nbase + 32] = acc2[r] + b1[nbase + 32];
        out2[gs + nbase + 48] = acc3[r] + b1[nbase + 48];
    }
}

// ---------------------------------------------------------------
// 6) classifier heads + log_softmax; one block (64 threads) per row
// ---------------------------------------------------------------
template <int NC>
__global__ void hann_head(const float* __restrict__ feat1, const float* __restrict__ feat2,
                          const int* __restrict__ idx1, const int* __restrict__ idx2,
                          const float* __restrict__ W, const float* __restrict__ b,
                          float* __restrict__ out) {
    __shared__ float tc[2 * D_];
    __shared__ float red[64];
    int t = threadIdx.x;
    int r1 = idx1[blockIdx.x];
    int r2 = idx2[blockIdx.x];
    for (int i = t; i < D_; i += 64) {
        tc[i]       = feat1[(size_t)r1 * D_ + i];
        tc[D_ + i]  = feat2[(size_t)r2 * D_ + i];
    }
    __syncthreads();

    float logit = -INFINITY;
    if (t < NC) {
        float acc = b[t];
        for (int k = 0; k < 2 * D_; ++k) acc += tc[k] * W[k * NC + t];
        logit = acc;
    }
    red[t] = logit;
    __syncthreads();
    for (int s = 32; s > 0; s >>= 1) {
        if (t < s) red[t] = fmaxf(red[t], red[t + s]);
        __syncthreads();
    }
    float mx = red[0];
    __syncthreads();
    red[t] = (t < NC) ? expf(logit - mx) : 0.0f;
    __syncthreads();
    for (int s = 32; s > 0; s >>= 1) {
        if (t < s) red[t] += red[t + s];
        __syncthreads();
    }
    float lse = logf(red[0]) + mx;
    if (t < NC) out[(size_t)blockIdx.x * NC + t] = logit - lse;
}

// ---------------------------------------------------------------
extern "C" void kernel_launch(void* const* d_in, const int* in_sizes, int n_in,
                              void* d_out, int out_size, void* d_ws, size_t ws_size,
                              hipStream_t stream) {
    (void)in_sizes; (void)n_in; (void)out_size; (void)ws_size;

    const int*   x    = (const int*)d_in[0];
    const int*   esrc = (const int*)d_in[1];
    const int*   edst = (const int*)d_in[2];
    const int*   hv   = (const int*)d_in[3];
    const int*   he   = (const int*)d_in[4];
    const int*   tidx = (const int*)d_in[5];
    const int*   tsen = (const int*)d_in[6];
    const int*   a1   = (const int*)d_in[7];
    const int*   a2   = (const int*)d_in[8];
    const float* emb  = (const float*)d_in[9];
    const float* Wc1  = (const float*)d_in[10];
    const float* bc1  = (const float*)d_in[11];
    const float* Wc2  = (const float*)d_in[12];
    const float* bc2  = (const float*)d_in[13];
    const float* W1   = (const float*)d_in[14];
    const float* b1   = (const float*)d_in[15];
    const float* W2   = (const float*)d_in[16];
    const float* b2   = (const float*)d_in[17];
    const float* W3   = (const float*)d_in[18];
    const float* b3   = (const float*)d_in[19];

    const size_t BIG = (size_t)S_ * L_ * D_;               // 16,777,216 floats
    float* bufA  = (float*)d_ws;
    float* bufB  = bufA + BIG;
    float* bufC  = bufB + BIG;
    float* dinv  = bufC + BIG;                             // S*L
    float* sents = dinv + (size_t)S_ * L_;                 // S*D

    float* prob_t = (float*)d_out;                         // [T, 40]
    float* prob_r = prob_t + (size_t)T_ * NT_;             // [A, 20]

    const int ELT_BLOCKS  = (int)(BIG / 256);              // 65536
    const int GEMM_STRIPS = (S_ * L_ / 16) * (D_ / 64);    // 16384

    // h0 = emb[x] + pe                      -> bufA
    hann_emb_pe<<<ELT_BLOCKS, 256, 0, stream>>>(x, emb, bufA);

    // degree / dinv (same for both layers)
    hann_deg_init<<<S_ * L_ / 256, 256, 0, stream>>>(dinv);
    hann_deg_accum<<<S_ * E_ / 256, 256, 0, stream>>>(edst, dinv);
    hann_dinv<<<S_ * L_ / 256, 256, 0, stream>>>(dinv);

    // layer 1: hW = h0 @ Wc1 -> bufB ; aggregate -> bufC ; bias+relu
    hann_gemm_f32<<<GEMM_STRIPS, 32, 0, stream>>>(bufA, Wc1, bufB, S_ * L_, D_, D_);
    hann_agg_init<<<ELT_BLOCKS, 256, 0, stream>>>(bufB, dinv, bufC);
    hann_agg_edges<<<S_ * E_, 256, 0, stream>>>(bufB, dinv, esrc, edst, bufC);
    hann_bias_relu<<<ELT_BLOCKS, 256, 0, stream>>>(bufC, bc1);

    // layer 2: hW = h1 @ Wc2 -> bufA ; aggregate -> bufB ; bias+relu
    hann_gemm_f32<<<GEMM_STRIPS, 32, 0, stream>>>(bufC, Wc2, bufA, S_ * L_, D_, D_);
    hann_agg_init<<<ELT_BLOCKS, 256, 0, stream>>>(bufA, dinv, bufB);
    hann_agg_edges<<<S_ * E_, 256, 0, stream>>>(bufA, dinv, esrc, edst, bufB);
    hann_bias_relu<<<ELT_BLOCKS, 256, 0, stream>>>(bufB, bc2);
    // bufB = outs (= h2 flattened)

    // sentence summaries and hypergraph max
    hann_sent_max<<<S_, D_, 0, stream>>>(bufB, sents);
    hann_hyper_max<<<N_, D_, 0, stream>>>(sents, he, bufC);   // bufC = m [N, D]

    // outs2 = copy(outs); scatter ffnn1 rows into it
    hipMemcpyAsync(bufA, bufB, BIG * sizeof(float), hipMemcpyDeviceToDevice, stream);
    hann_ffnn1_wmma<<<(N_ / 16) * (D_ / 64), 32, 0, stream>>>(bufB, bufC, hv, W1, b1, bufA);

    // heads
    hann_head<NT_><<<T_, 64, 0, stream>>>(bufA, sents, tidx, tsen, W2, b2, prob_t);
    hann_head<NR_><<<A_, 64, 0, stream>>>(bufA, bufA,  a1,   a2,   W3, b3, prob_r);
}